// ProposalFilter_3307124818720
// MI455X (gfx1250) — compile-verified
//
#include <hip/hip_runtime.h>
#include <hip/hip_bf16.h>
#include <stdint.h>

#define MAXP        300
#define SORT_CAP    4096          // next pow2 >= N (N = 4000)
#define BLOCK       1024          // 32 wave32 waves on one WGP
#define IOU_THR     0.5f
#define MIN_SIZE    16.0f
#define SCORE_THR   0.5f

// -------- gfx1250 async global->LDS path (guarded; fallback = plain copy) ----
#if defined(__has_builtin)
#  if __has_builtin(__builtin_amdgcn_global_load_async_to_lds_b128)
#    define USE_ASYNC_LDS 1
#  endif
#endif
#ifndef USE_ASYNC_LDS
#  define USE_ASYNC_LDS 0
#endif

typedef int v4i __attribute__((vector_size(16)));
typedef __attribute__((address_space(1))) v4i* gptr_v4i;
typedef __attribute__((address_space(3))) v4i* lptr_v4i;

__device__ __forceinline__ void async_copy16(const void* gsrc, void* lds_dst) {
#if USE_ASYNC_LDS
  // addrspace punning via integer cast: global flat addr == addrspace(1) addr;
  // LDS offset == low 32 bits of flat addr (aperture layout).
  __builtin_amdgcn_global_load_async_to_lds_b128(
      (gptr_v4i)(uintptr_t)gsrc,
      (lptr_v4i)(unsigned)(uintptr_t)lds_dst,
      0, 0);
#else
  *(float4*)lds_dst = *(const float4*)gsrc;
#endif
}

__device__ __forceinline__ void async_wait_all() {
#if USE_ASYNC_LDS
#  if defined(__has_builtin) && __has_builtin(__builtin_amdgcn_s_wait_asynccnt)
  __builtin_amdgcn_s_wait_asynccnt(0);
#  else
  asm volatile("s_wait_asynccnt 0" ::: "memory");
#  endif
#endif
}

// comparator: should element a come before element b? (score desc, index asc = stable argsort)
__device__ __forceinline__ bool before(float ka, int ia, float kb, int ib) {
  return (ka > kb) || (ka == kb && ia < ib);
}

__global__ __launch_bounds__(BLOCK)
void proposal_filter_kernel(const float* __restrict__ props,
                            const float* __restrict__ scores,
                            float* __restrict__ out, int N) {
  __shared__ float    s_key[SORT_CAP];
  __shared__ int      s_idx[SORT_CAP];
  __shared__ float4   s_box[SORT_CAP];   // sorted boxes, resident in 320KB WGP LDS
  __shared__ unsigned s_keep[SORT_CAP / 32];
  __shared__ int      s_wpre[SORT_CAP / 32 + 1];
  __shared__ int      s_nv;

  const int b   = blockIdx.x;
  const int tid = threadIdx.x;
  const float NEG_INF = -__builtin_inff();

  const float* bprops  = props  + (size_t)b * N * 4;
  const float* bscores = scores + (size_t)b * N * 2;
  float*       bout    = out    + (size_t)b * MAXP * 4;

  if (tid == 0) s_nv = SORT_CAP;

  // ---- Phase 1: filter + build (key,idx) ----------------------------------
  for (int t = tid; t < SORT_CAP; t += BLOCK) {
    float key = NEG_INF;
    if (t < N) {
      float fg = bscores[2 * t + 1];
      const float* bp = bprops + 4 * t;
      float w = bp[2] - bp[0];
      float h = bp[3] - bp[1];
      if (fg > SCORE_THR && w >= MIN_SIZE && h >= MIN_SIZE) key = fg;
    }
    s_key[t] = key;
    s_idx[t] = t;
  }

  // ---- Phase 2: bitonic sort (score desc, index asc) ----------------------
  for (int k = 2; k <= SORT_CAP; k <<= 1) {
    for (int j = k >> 1; j > 0; j >>= 1) {
      __syncthreads();
      for (int t = tid; t < SORT_CAP; t += BLOCK) {
        int l = t ^ j;
        if (l > t) {
          float ka = s_key[t], kb = s_key[l];
          int   ia = s_idx[t], ib = s_idx[l];
          bool up = ((t & k) == 0);
          bool sw = up ? before(kb, ib, ka, ia)   // ascending-in-comparator block
                       : before(ka, ia, kb, ib);  // descending block
          if (sw) {
            s_key[t] = kb; s_key[l] = ka;
            s_idx[t] = ib; s_idx[l] = ia;
          }
        }
      }
    }
  }
  __syncthreads();

  // ---- Phase 3: n_valid, keep-bitmask init, async box gather --------------
  for (int t = tid; t < SORT_CAP; t += BLOCK)
    if (!(s_key[t] > 0.0f)) atomicMin(&s_nv, t);   // sorted: first invalid index

  for (int w = tid; w < SORT_CAP / 32; w += BLOCK) {
    unsigned m = 0u;
    for (int r = 0; r < 32; ++r)
      if (s_key[w * 32 + r] > 0.0f) m |= (1u << r);
    s_keep[w] = m;
  }

  for (int t = tid; t < SORT_CAP; t += BLOCK) {
    int gi = s_idx[t];
    if (gi >= N) gi = N - 1;                        // padding entries: clamp (never output)
    async_copy16(bprops + 4 * gi, &s_box[t]);       // global_load_async_to_lds_b128
  }
  async_wait_all();                                 // s_wait_asynccnt 0
  __syncthreads();

  const int n = s_nv;

  // ---- Phase 4: greedy sequential NMS (uniform control flow) --------------
  int kept = 0;
  for (int i = 0; i < n; ++i) {
    __syncthreads();                                // make prior clears visible
    unsigned wbits = s_keep[i >> 5];
    if (!((wbits >> (i & 31)) & 1u)) continue;
    ++kept;
    if (kept >= MAXP) break;                        // first 300 kept are final
    float4 bi = s_box[i];
    float areai = (bi.z - bi.x) * (bi.w - bi.y);
    for (int j = i + 1 + tid; j < n; j += BLOCK) {
      float4 bj = s_box[j];
      float xx1 = fmaxf(bi.x, bj.x), yy1 = fmaxf(bi.y, bj.y);
      float xx2 = fminf(bi.z, bj.z), yy2 = fminf(bi.w, bj.w);
      float inter = fmaxf(xx2 - xx1, 0.0f) * fmaxf(yy2 - yy1, 0.0f);
      float areaj = (bj.z - bj.x) * (bj.w - bj.y);
      float uni   = areai + areaj - inter;
      if (inter / uni >= IOU_THR)
        atomicAnd(&s_keep[j >> 5], ~(1u << (j & 31)));
    }
  }
  __syncthreads();

  // ---- Phase 5: rank by prefix popcount, compact + zero-fill --------------
  if (tid == 0) {
    int acc = 0;
    for (int w = 0; w < SORT_CAP / 32; ++w) { s_wpre[w] = acc; acc += __popc(s_keep[w]); }
    s_wpre[SORT_CAP / 32] = acc;
  }
  __syncthreads();

  int total = s_wpre[SORT_CAP / 32];
  int kcap  = total < MAXP ? total : MAXP;

  for (int e = kcap * 4 + tid; e < MAXP * 4; e += BLOCK) bout[e] = 0.0f;

  for (int j = tid; j < n; j += BLOCK) {
    unsigned wb = s_keep[j >> 5];
    if ((wb >> (j & 31)) & 1u) {
      int rank = s_wpre[j >> 5] + __popc(wb & ((1u << (j & 31)) - 1u));
      if (rank < MAXP) {
        float4 bb = s_box[j];
        bout[rank * 4 + 0] = bb.x;
        bout[rank * 4 + 1] = bb.y;
        bout[rank * 4 + 2] = bb.z;
        bout[rank * 4 + 3] = bb.w;
      }
    }
  }
}

extern "C" void kernel_launch(void* const* d_in, const int* in_sizes, int n_in,
                              void* d_out, int out_size, void* d_ws, size_t ws_size,
                              hipStream_t stream) {
  const float* props  = (const float*)d_in[0];   // (B, N, 4) f32
  const float* scores = (const float*)d_in[1];   // (B, N, 2) f32
  float* out = (float*)d_out;                    // (B, 300, 4) f32

  int B = out_size / (MAXP * 4);
  int N = in_sizes[0] / (4 * B);

  proposal_filter_kernel<<<B, BLOCK, 0, stream>>>(props, scores, out, N);
}